// metric_chamferDist_28999619182831
// MI455X (gfx1250) — compile-verified
//
#include <hip/hip_runtime.h>

typedef __attribute__((ext_vector_type(2))) float v2f;
typedef __attribute__((ext_vector_type(8))) float v8f;

#define GS      128
#define NPTS    (GS * GS)        // 16384 points per grid
#define RES     0.25f
#define NGRIDS  16
#define TILES   (NPTS / 16)      // 1024 16-wide tiles
#define WPB     8                // waves per block
#define THREADS (WPB * 32)

// One wave computes, for one (grid, direction, 16-row A tile), the per-row
// min over all 16384 B points of max(|a-b|^2, 0), using
// |a-b|^2 = |a|^2 + (|b|^2 - 2 a.b), with a.b done on the matrix pipe via
// V_WMMA_F32_16X16X4_F32 (K=3 padded to 4).
__global__ void __launch_bounds__(THREADS)
chamfer_main(const float* __restrict__ pred, const float* __restrict__ gt,
             float* __restrict__ acc)
{
    __shared__ float zB[NPTS];           // 64 KB: B-grid z values
    __shared__ float wsum[WPB];

    const int tid  = threadIdx.x;
    const int lane = tid & 31;
    const int wave = tid >> 5;
    const int half = lane >> 4;          // 0: lanes 0-15, 1: lanes 16-31
    const int l15  = lane & 15;

    const int wg0  = blockIdx.x * WPB;   // 8 consecutive waves share (g, dir)
    const int g    = wg0 >> 11;          // 2048 waves per grid
    const int dir  = (wg0 >> 10) & 1;    // 0: A=gt,B=pred ; 1: A=pred,B=gt
    const int tile = (wg0 & 1023) + wave;
    const int m0   = tile << 4;

    const float* Ag = (dir == 0 ? gt   : pred) + g * NPTS;
    const float* Bg = (dir == 0 ? pred : gt)   + g * NPTS;

    // Stage B z-values into LDS (64 iterations, fully uniform -> EXEC all 1s)
    for (int i = tid; i < NPTS; i += THREADS)
        zB[i] = Bg[i];
    __syncthreads();

    // A fragment, 16x4 f32 layout: lanes 0-15 hold K={0,1}=(x,y),
    // lanes 16-31 hold K={2,3}=(z,0), M = lane%16.
    const int   arow = m0 + l15;
    const float xa   = (float)(arow >> 7) * RES;
    const float ya   = (float)(arow & (GS - 1)) * RES;
    const float za   = Ag[arow];
    v2f afrag;
    afrag.x = half ? za   : xa;
    afrag.y = half ? 0.0f : ya;

    v8f minv;
#pragma unroll
    for (int i = 0; i < 8; ++i) minv[i] = 3.0e38f;

    const v8f czero = {};

    for (int t = 0; t < TILES; ++t) {
        const int   col = (t << 4) + l15;
        const float zb  = zB[col];
        const float xb  = (float)(col >> 7) * RES;
        const float yb  = (float)(col & (GS - 1)) * RES;
        const float rb2 = xb * xb + yb * yb + zb * zb;

        // B fragment, 4x16 f32 layout: lanes 0-15 hold K={0,1},
        // lanes 16-31 hold K={2,3}, N = lane%16.
        v2f bfrag;
        bfrag.x = half ? zb   : xb;
        bfrag.y = half ? 0.0f : yb;

        // dot[(M,N)] = a_M . b_N on the matrix pipe
        v8f dot = __builtin_amdgcn_wmma_f32_16x16x4_f32(
            false, afrag, false, bfrag, (short)0, czero, false, false);

#pragma unroll
        for (int i = 0; i < 8; ++i) {
            float cand = fmaf(-2.0f, dot[i], rb2);   // |b|^2 - 2 a.b
            minv[i] = fminf(minv[i], cand);
        }
    }

    // Min-reduce across the 16 lanes of each half-wave (over columns N).
#pragma unroll
    for (int m = 1; m < 16; m <<= 1) {
#pragma unroll
        for (int i = 0; i < 8; ++i)
            minv[i] = fminf(minv[i], __shfl_xor(minv[i], m, 32));
    }

    // Lanes 0 and 16 each finalize 8 rows: add |a|^2, clamp, accumulate.
    float part = 0.0f;
    if (l15 == 0) {
#pragma unroll
        for (int v = 0; v < 8; ++v) {
            const int   m   = m0 + v + (half << 3);
            const float x   = (float)(m >> 7) * RES;
            const float y   = (float)(m & (GS - 1)) * RES;
            const float z   = Ag[m];
            const float ra2 = x * x + y * y + z * z;
            part += fmaxf(ra2 + minv[v], 0.0f);
        }
    }
    part += __shfl_xor(part, 16, 32);    // lane0 + lane16
    if (lane == 0) wsum[wave] = part;
    __syncthreads();

    if (tid == 0) {
        float s = 0.0f;
#pragma unroll
        for (int w = 0; w < WPB; ++w) s += wsum[w];
        atomicAdd(acc, s);
    }
}

__global__ void chamfer_zero(float* acc) { *acc = 0.0f; }

__global__ void chamfer_finalize(const float* __restrict__ acc, float* __restrict__ out)
{
    *out = *acc * (1.0f / (float)NGRIDS);
}

extern "C" void kernel_launch(void* const* d_in, const int* in_sizes, int n_in,
                              void* d_out, int out_size, void* d_ws, size_t ws_size,
                              hipStream_t stream)
{
    const float* pred = (const float*)d_in[0];   // inputs  (1,1,512,512) -> 16 pred grids
    const float* gt   = (const float*)d_in[1];   // targets (1,512,512)   -> 16 gt grids
    float* acc = (float*)d_ws;
    float* out = (float*)d_out;

    chamfer_zero<<<1, 1, 0, stream>>>(acc);
    const int nblocks = NGRIDS * 2 * TILES / WPB;   // 4096 blocks of 8 waves
    chamfer_main<<<nblocks, THREADS, 0, stream>>>(pred, gt, acc);
    chamfer_finalize<<<1, 1, 0, stream>>>(acc, out);
}